// TripletMarginLossWithNegativeMining_39530878992715
// MI455X (gfx1250) — compile-verified
//
#include <hip/hip_runtime.h>

#define BB 512
#define DD 128
#define KK 256
#define MARGIN_F 1.0f

typedef __attribute__((ext_vector_type(2))) float v2f;
typedef __attribute__((ext_vector_type(8))) float v8f;

// ---------------------------------------------------------------------------
// Kernel 1: per-row stats (||a||^2, ||n||^2, pos_dist) + zero the count array
// ---------------------------------------------------------------------------
__global__ void tml_stats_kernel(const float* __restrict__ a,
                                 const float* __restrict__ p,
                                 const float* __restrict__ n,
                                 float* __restrict__ a2,
                                 float* __restrict__ n2,
                                 float* __restrict__ pd,
                                 int* __restrict__ count) {
    int i = blockIdx.x * blockDim.x + threadIdx.x;
    if (i >= BB) return;
    const float4* ar = reinterpret_cast<const float4*>(a + (size_t)i * DD);
    const float4* pr = reinterpret_cast<const float4*>(p + (size_t)i * DD);
    const float4* nr = reinterpret_cast<const float4*>(n + (size_t)i * DD);
    float sa = 0.f, sn = 0.f, sp = 0.f;
#pragma unroll 4
    for (int q = 0; q < DD / 4; ++q) {
        float4 av = ar[q], pv = pr[q], nv = nr[q];
        sa += av.x * av.x + av.y * av.y + av.z * av.z + av.w * av.w;
        sn += nv.x * nv.x + nv.y * nv.y + nv.z * nv.z + nv.w * nv.w;
        float dx = av.x - pv.x, dy = av.y - pv.y, dz = av.z - pv.z, dw = av.w - pv.w;
        sp += dx * dx + dy * dy + dz * dz + dw * dw;
    }
    a2[i] = sa;
    n2[i] = sn;
    pd[i] = sp;
    count[i] = 0;   // zero accumulator every call (harness does not re-poison)
}

// ---------------------------------------------------------------------------
// Kernel 2: WMMA fp32 GEMM G = A * N^T, emit neg_dist = a2 + n2 - 2G,
//           and accumulate sum of relu(margin + pd[col] - neg_dist)
//           One wave per 16x16 tile; 32 k-steps of V_WMMA_F32_16X16X4_F32.
// ---------------------------------------------------------------------------
__global__ void tml_gemm_dist_kernel(const float* __restrict__ a,
                                     const float* __restrict__ n,
                                     const float* __restrict__ a2,
                                     const float* __restrict__ n2,
                                     const float* __restrict__ pd,
                                     float* __restrict__ nd,
                                     float* __restrict__ lossPart) {
    __shared__ float sred[256];
    const int lane   = threadIdx.x & 31;
    const int waveId = blockIdx.x * (blockDim.x >> 5) + (threadIdx.x >> 5);
    const int ti = waveId >> 5;      // tile row   (0..31)
    const int tj = waveId & 31;      // tile col   (0..31)
    const int half = lane >> 4;      // 0: K pair {0,1}, 1: K pair {2,3}
    const int idx  = lane & 15;      // M (for A) / N (for B) index in tile

    const float* arow = a + (size_t)(ti * 16 + idx) * DD;
    const float* nrow = n + (size_t)(tj * 16 + idx) * DD;

    v8f c = {};
    for (int kb = 0; kb < DD; kb += 4) {
        const int ko = kb + 2 * half;
        float2 af = *reinterpret_cast<const float2*>(arow + ko);
        float2 bf = *reinterpret_cast<const float2*>(nrow + ko);
        v2f av = {af.x, af.y};
        v2f bv = {bf.x, bf.y};
        // D = A(16x4 f32) * B(4x16 f32) + C, full fp32 matrix pipe
        c = __builtin_amdgcn_wmma_f32_16x16x4_f32(
                /*neg_a=*/false, av, /*neg_b=*/false, bv,
                /*c_mod=*/(short)0, c, /*reuse_a=*/false, /*reuse_b=*/false);
    }

    const int col = tj * 16 + idx;
    const float n2c = n2[col];
    const float pdc = pd[col];     // reference: pos_dist[None,:] -> indexed by COLUMN
    float lsum = 0.f;
#pragma unroll
    for (int v = 0; v < 8; ++v) {
        const int row = ti * 16 + v + 8 * half;   // C/D layout: VGPR v, lanes 16-31 hold M=v+8
        const float dist = a2[row] + n2c - 2.0f * c[v];
        nd[(size_t)row * BB + col] = dist;
        const float t = MARGIN_F + pdc - dist;
        lsum += (t > 0.f) ? t : 0.f;
    }

    // deterministic block tree reduction -> per-block partial
    sred[threadIdx.x] = lsum;
    __syncthreads();
    for (int s = blockDim.x >> 1; s > 0; s >>= 1) {
        if ((int)threadIdx.x < s) sred[threadIdx.x] += sred[threadIdx.x + s];
        __syncthreads();
    }
    if (threadIdx.x == 0) lossPart[blockIdx.x] = sred[0];
}

// ---------------------------------------------------------------------------
// Kernel 3: per-row rank selection of the k smallest (exact jax.lax.top_k tie
//           semantics: stable by index) -> integer histogram of selected cols
// ---------------------------------------------------------------------------
__global__ void tml_topk_count_kernel(const float* __restrict__ nd,
                                      int* __restrict__ count) {
    __shared__ float row[BB];
    const int i = blockIdx.x;
    const int j = threadIdx.x;
    row[j] = nd[(size_t)i * BB + j];
    __syncthreads();
    const float vj = row[j];
    int rank = 0;
#pragma unroll 8
    for (int q = 0; q < BB; ++q) {
        const float vq = row[q];
        rank += (vq < vj) || (vq == vj && q < j);
    }
    if (rank < KK) atomicAdd(&count[j], 1);   // integer atomic: exact & deterministic
}

// ---------------------------------------------------------------------------
// Kernel 4: hnl row sums: sum_j count[j] * relu(margin + pd[i] - nd[i,j])
// ---------------------------------------------------------------------------
__global__ void tml_hnl_kernel(const float* __restrict__ nd,
                               const int* __restrict__ count,
                               const float* __restrict__ pd,
                               float* __restrict__ hnlPart) {
    __shared__ float sred[BB];
    const int i = blockIdx.x;
    const int j = threadIdx.x;
    const float dist = nd[(size_t)i * BB + j];
    float t = MARGIN_F + pd[i] - dist;        // reference: pos_dist[:,None] -> ROW index
    t = (t > 0.f) ? t : 0.f;
    sred[j] = t * (float)count[j];
    __syncthreads();
    for (int s = BB >> 1; s > 0; s >>= 1) {
        if (j < s) sred[j] += sred[j + s];
        __syncthreads();
    }
    if (j == 0) hnlPart[i] = sred[0];
}

// ---------------------------------------------------------------------------
// Kernel 5: combine partials into the final scalar (fixed order -> bit stable)
// ---------------------------------------------------------------------------
__global__ void tml_final_kernel(const float* __restrict__ lossPart,
                                 const float* __restrict__ hnlPart,
                                 float* __restrict__ out) {
    float ls = 0.f, hs = 0.f;
    for (int q = 0; q < 128; ++q) ls += lossPart[q];
    for (int q = 0; q < BB; ++q) hs += hnlPart[q];
    // mean(loss) + mean(mean(hnl,axis=1)) = lossSum/B^2 + hnlSum/(B*k*B)
    out[0] = ls / ((float)BB * (float)BB) +
             hs / ((float)BB * (float)BB * (float)KK);
}

extern "C" void kernel_launch(void* const* d_in, const int* in_sizes, int n_in,
                              void* d_out, int out_size, void* d_ws, size_t ws_size,
                              hipStream_t stream) {
    const float* anchor   = (const float*)d_in[0];
    const float* positive = (const float*)d_in[1];
    const float* negative = (const float*)d_in[2];
    float* out = (float*)d_out;

    // workspace layout (≈1.01 MiB total)
    float* nd       = (float*)d_ws;          // [512*512] neg_dist
    float* a2       = nd + (size_t)BB * BB;  // [512]
    float* n2       = a2 + BB;               // [512]
    float* pd       = n2 + BB;               // [512]
    int*   count    = (int*)(pd + BB);       // [512]
    float* lossPart = (float*)(count + BB);  // [128]
    float* hnlPart  = lossPart + 128;        // [512]

    tml_stats_kernel<<<2, 256, 0, stream>>>(anchor, positive, negative,
                                            a2, n2, pd, count);
    tml_gemm_dist_kernel<<<128, 256, 0, stream>>>(anchor, negative,
                                                  a2, n2, pd, nd, lossPart);
    tml_topk_count_kernel<<<BB, BB, 0, stream>>>(nd, count);
    tml_hnl_kernel<<<BB, BB, 0, stream>>>(nd, count, pd, hnlPart);
    tml_final_kernel<<<1, 1, 0, stream>>>(lossPart, hnlPart, out);
}